// SqueezeExcitationBlock_61168924230407
// MI455X (gfx1250) — compile-verified
//
#include <hip/hip_runtime.h>

// Squeeze-Excitation, fused single kernel for MI455X (gfx1250, wave32).
//
// Roofline: ~0.4 GFLOP vs ~1.5 GB traffic -> pure bandwidth problem.
//   phase 1: read x (512 MiB), per-channel mean
//   phase 2: 64->16->64 MLP via fp32 WMMA (V_WMMA_F32_16X16X4_F32), redundant per wave
//   phase 3: re-read x (reversed order -> partial 192MB-L2 hits), scale, NT store
// One workgroup (1024 thr = 32 waves) per batch; each wave owns 2 channels.

typedef __attribute__((ext_vector_type(2))) float v2f;
typedef __attribute__((ext_vector_type(4))) float v4f;
typedef __attribute__((ext_vector_type(8))) float v8f;

#define B_    128
#define C_    64
#define T_    16384
#define RED_  16
#define WAVES 32
#define CHUNKS (T_ / 4 / 32)  // 128 float4 chunks per lane per channel

__device__ __forceinline__ float sigmoidf_(float v) {
  return 1.0f / (1.0f + __expf(-v));
}

__global__ __launch_bounds__(1024, 1)
void se_fused_kernel(const float* __restrict__ x,
                     const float* __restrict__ w1, const float* __restrict__ b1,
                     const float* __restrict__ w2, const float* __restrict__ b2,
                     float* __restrict__ out)
{
  __shared__ float u_sh[C_];               // per-batch channel means
  __shared__ float h_sh[WAVES * RED_];     // per-wave hidden activations
  __shared__ float s_sh[C_];               // sigmoid gates (benign-race identical writes)

  const int tid  = threadIdx.x;
  const int lane = tid & 31;
  const int wave = tid >> 5;
  const int b    = blockIdx.x;

  // ---------------- Phase 1: mean over T (2 channels per wave) ----------------
  for (int cc = 0; cc < 2; ++cc) {
    const int c = wave * 2 + cc;
    const v4f* row = (const v4f*)(x + ((size_t)b * C_ + c) * T_);
    float sum = 0.0f;
    for (int i = 0; i < CHUNKS; ++i) {
      v4f v = row[i * 32 + lane];          // coalesced global_load_b128
      sum += (v.x + v.y) + (v.z + v.w);
    }
#pragma unroll
    for (int m = 16; m >= 1; m >>= 1) sum += __shfl_xor(sum, m, 32);
    if (lane == 0) u_sh[c] = sum * (1.0f / (float)T_);
  }
  __syncthreads();

  // ---------------- Phase 2: excitation MLP via fp32 WMMA ----------------
  // All waves compute redundantly so EXEC is all-1s at every WMMA.
  // fp32 A(16x4) layout (ISA 7.12.2): lane L holds M=L&15; VGPR0 = K {0|2},
  // VGPR1 = K {1|3} selected by lane half. B(4x16) mirrored: lane = column N.
  // C/D(16x16): VGPR r -> M=r (lanes 0-15) / M=r+8 (lanes 16-31), N = lane&15.
  const int  mrow = lane & 15;   // A: row M, B: column N (numerically the same)
  const int  khal = lane >> 4;   // selects K pair {0,1} vs {2,3}
  const bool col0 = (mrow == 0); // only column 0 of B carries the vector

  // GEMM1: h_pre[16] = w1[16,64] @ u[64]; result lands in D column N==0.
  // Branch-free B build: unconditional LDS load in all lanes + v_cndmask select
  // (avoids exec-masked ds_load / s_and_saveexec churn seen in the prior asm).
  v8f acc = {};
  for (int k0 = 0; k0 < C_; k0 += 4) {
    v2f a, bv;
    a.x = w1[mrow * C_ + k0 + 2 * khal + 0];
    a.y = w1[mrow * C_ + k0 + 2 * khal + 1];
    const float ux = u_sh[k0 + 2 * khal + 0];
    const float uy = u_sh[k0 + 2 * khal + 1];
    bv.x = col0 ? ux : 0.0f;
    bv.y = col0 ? uy : 0.0f;
    acc = __builtin_amdgcn_wmma_f32_16x16x4_f32(false, a, false, bv,
                                                (short)0, acc, false, false);
  }
  if (lane == 0) {
#pragma unroll
    for (int r = 0; r < 8; ++r) {
      float hv = acc[r] + b1[r];
      h_sh[wave * RED_ + r] = hv > 0.0f ? hv : 0.0f;
    }
  }
  if (lane == 16) {
#pragma unroll
    for (int r = 0; r < 8; ++r) {
      float hv = acc[r] + b1[8 + r];
      h_sh[wave * RED_ + 8 + r] = hv > 0.0f ? hv : 0.0f;
    }
  }
  __syncthreads();

  // GEMM2: s_pre[64] = w2[64,16] @ h[16]; 4 M-tiles of 16, K=16 in 4 steps.
  for (int mt = 0; mt < 4; ++mt) {
    v8f acc2 = {};
#pragma unroll
    for (int k0 = 0; k0 < RED_; k0 += 4) {
      v2f a, bv;
      a.x = w2[(mt * 16 + mrow) * RED_ + k0 + 2 * khal + 0];
      a.y = w2[(mt * 16 + mrow) * RED_ + k0 + 2 * khal + 1];
      const float hx = h_sh[wave * RED_ + k0 + 2 * khal + 0];
      const float hy = h_sh[wave * RED_ + k0 + 2 * khal + 1];
      bv.x = col0 ? hx : 0.0f;
      bv.y = col0 ? hy : 0.0f;
      acc2 = __builtin_amdgcn_wmma_f32_16x16x4_f32(false, a, false, bv,
                                                   (short)0, acc2, false, false);
    }
    if (lane == 0) {
#pragma unroll
      for (int r = 0; r < 8; ++r)
        s_sh[mt * 16 + r] = sigmoidf_(acc2[r] + b2[mt * 16 + r]);
    }
    if (lane == 16) {
#pragma unroll
      for (int r = 0; r < 8; ++r)
        s_sh[mt * 16 + 8 + r] = sigmoidf_(acc2[r] + b2[mt * 16 + 8 + r]);
    }
  }
  __syncthreads();

  // Emit the s output (second tuple element, concatenated after x_recal).
  if (tid < C_)
    out[(size_t)B_ * C_ * T_ + (size_t)b * C_ + tid] = s_sh[tid];

  // ---------------- Phase 3: rescale, REVERSED order for L2 reuse ----------------
  // Tail of phase-1 data is hottest in L2; walk backwards so re-reads hit.
  // Loads are last-use and stores never re-read -> non-temporal on both.
  for (int cc = 1; cc >= 0; --cc) {
    const int c = wave * 2 + cc;
    const float sc = s_sh[c];
    const v4f* row  = (const v4f*)(x   + ((size_t)b * C_ + c) * T_);
    v4f*       orow = (v4f*)      (out + ((size_t)b * C_ + c) * T_);
    for (int i = CHUNKS - 1; i >= 0; --i) {
      const int idx = i * 32 + lane;
      v4f v = __builtin_nontemporal_load(&row[idx]);
      v *= sc;
      __builtin_nontemporal_store(v, &orow[idx]);
    }
  }
}

extern "C" void kernel_launch(void* const* d_in, const int* in_sizes, int n_in,
                              void* d_out, int out_size, void* d_ws, size_t ws_size,
                              hipStream_t stream) {
  (void)in_sizes; (void)n_in; (void)out_size; (void)d_ws; (void)ws_size;
  const float* x  = (const float*)d_in[0];
  const float* w1 = (const float*)d_in[1];
  const float* b1 = (const float*)d_in[2];
  const float* w2 = (const float*)d_in[3];
  const float* b2 = (const float*)d_in[4];
  float* out = (float*)d_out;
  se_fused_kernel<<<B_, 1024, 0, stream>>>(x, w1, b1, w2, b2, out);
}